// ExtendedKalmanFilter_39178691674375
// MI455X (gfx1250) — compile-verified
//
#include <hip/hip_runtime.h>
#include <hip/hip_bf16.h>

typedef __attribute__((ext_vector_type(2))) float v2f;
typedef __attribute__((ext_vector_type(8))) float v8f;

#if __has_builtin(__builtin_amdgcn_wmma_f32_16x16x4_f32)
#define EKF_HAVE_WMMA_F32 1
#else
#define EKF_HAVE_WMMA_F32 0
#endif

#define BM 64
#define BN 128
#define BK 16
#define APAD 20   // As row stride (floats): rows 8B-aligned, conflict-free b64 frag loads
#define BKP 18    // Bs row stride (floats), n-major: frag pair contiguous -> ds_load_b64

// Generic fp32 GEMM on the CDNA5 matrix pipe (V_WMMA_F32_16X16X4_F32):
//   C[M,N] = A[M,K] * op(B) (+ addsrc)   op(B) = B[K,N] or B[N,K]^T
// KSCALE: A[m,k] *= kscale[k] (for B * diag(s) * B^T).
// Block = 256 threads = 8 wave32s (2x4); each wave owns 32x32 (2x2 WMMA tiles),
// so every LDS fragment feeds two WMMAs. Global->LDS is register double-buffered:
// tile kb+BK is prefetched into VGPRs while tile kb's 16 WMMAs execute.
// A tile LDS is row-major (frag pair contiguous); B tile LDS is n-major
// (Bs[n][k] = op(B)[k][n]) so B frag pairs are contiguous too -> no repack movs.
// Requires M % 64 == 0, N % 128 == 0, K % 16 == 0 (true: 1024/4096).
template <bool TRANSB, bool KSCALE, bool ADDSRC>
__global__ __launch_bounds__(256) void ekf_gemm_f32_wmma(
    const float* __restrict__ A, int lda,
    const float* __restrict__ Bm, int ldb,
    const float* __restrict__ kscale,
    const float* __restrict__ addsrc,
    float* __restrict__ C, int ldc,
    int M, int N, int K) {
  __shared__ float As[BM][APAD];
  __shared__ float Bs[BN][BKP];

  const int tid  = threadIdx.x;
  const int wave = tid >> 5;       // 0..7 (wave32)
  const int lane = tid & 31;
  const int half = lane >> 4;      // lane-half selects matrix half (ISA layout)
  const int lr   = lane & 15;
  const int m0   = blockIdx.y * BM;
  const int n0   = blockIdx.x * BN;
  const int wr   = wave >> 2;      // 0..1 -> 32-row band
  const int wc   = wave & 3;       // 0..3 -> 32-col band

  v8f acc00 = {}, acc01 = {}, acc10 = {}, acc11 = {};

  // A-tile coords: 64 rows x 16 cols, one float4 per thread
  const int ar = tid >> 2;            // 0..63
  const int ac = (tid & 3) << 2;      // 0,4,8,12
  // B-tile coords
  const int bnr = tid >> 5;           // non-trans: k row 0..7 (x2 halves)
  const int bnc = (tid & 31) << 2;    // non-trans: n col 0..124
  const int btn = tid >> 2;           // trans: n row 0..63 (x2 halves)
  const int btc = (tid & 3) << 2;     // trans: k col 0,4,8,12

  float4 pa, pks, pb0, pb1;

  auto prefetch = [&](int kb) {
    pa = *(const float4*)(A + (size_t)(m0 + ar) * lda + kb + ac);
    if (KSCALE) pks = *(const float4*)(kscale + kb + ac);
    if (!TRANSB) {
      pb0 = *(const float4*)(Bm + (size_t)(kb + bnr) * ldb + n0 + bnc);
      pb1 = *(const float4*)(Bm + (size_t)(kb + bnr + 8) * ldb + n0 + bnc);
    } else {
      pb0 = *(const float4*)(Bm + (size_t)(n0 + btn) * ldb + kb + btc);
      pb1 = *(const float4*)(Bm + (size_t)(n0 + btn + 64) * ldb + kb + btc);
    }
  };

  auto commit = [&]() {
    float4 v = pa;
    if (KSCALE) { v.x *= pks.x; v.y *= pks.y; v.z *= pks.z; v.w *= pks.w; }
    *(float4*)&As[ar][ac] = v;
    if (!TRANSB) {
      // Bs[n][k] = Bm[kb+k][n0+n]
      Bs[bnc + 0][bnr] = pb0.x; Bs[bnc + 1][bnr] = pb0.y;
      Bs[bnc + 2][bnr] = pb0.z; Bs[bnc + 3][bnr] = pb0.w;
      Bs[bnc + 0][bnr + 8] = pb1.x; Bs[bnc + 1][bnr + 8] = pb1.y;
      Bs[bnc + 2][bnr + 8] = pb1.z; Bs[bnc + 3][bnr + 8] = pb1.w;
    } else {
      // Bs[n][k] = Bm[n0+n][kb+k]  (contiguous along k)
      *(v2f*)&Bs[btn][btc]          = *(const v2f*)&pb0.x;
      *(v2f*)&Bs[btn][btc + 2]      = *(const v2f*)&pb0.z;
      *(v2f*)&Bs[btn + 64][btc]     = *(const v2f*)&pb1.x;
      *(v2f*)&Bs[btn + 64][btc + 2] = *(const v2f*)&pb1.z;
    }
  };

  prefetch(0);

  const float* aRow0 = &As[wr * 32 +  0 + lr][0];
  const float* aRow1 = &As[wr * 32 + 16 + lr][0];
  const float* bRow0 = &Bs[wc * 32 +  0 + lr][0];
  const float* bRow1 = &Bs[wc * 32 + 16 + lr][0];

  for (int kb = 0; kb < K; kb += BK) {
    commit();
    __syncthreads();
    if (kb + BK < K) prefetch(kb + BK);   // overlap HBM latency with WMMAs

    #pragma unroll
    for (int k = 0; k < BK; k += 4) {
      const int kk = k + half * 2;
#if EKF_HAVE_WMMA_F32
      v2f a0 = *(const v2f*)(aRow0 + kk);
      v2f a1 = *(const v2f*)(aRow1 + kk);
      v2f b0 = *(const v2f*)(bRow0 + kk);
      v2f b1 = *(const v2f*)(bRow1 + kk);
      acc00 = __builtin_amdgcn_wmma_f32_16x16x4_f32(false, a0, false, b0, (short)0, acc00, false, false);
      acc01 = __builtin_amdgcn_wmma_f32_16x16x4_f32(false, a0, false, b1, (short)0, acc01, false, false);
      acc10 = __builtin_amdgcn_wmma_f32_16x16x4_f32(false, a1, false, b0, (short)0, acc10, false, false);
      acc11 = __builtin_amdgcn_wmma_f32_16x16x4_f32(false, a1, false, b1, (short)0, acc11, false, false);
#else
      (void)kk;
      #pragma unroll
      for (int v = 0; v < 8; ++v) {
        float s00 = acc00[v], s01 = acc01[v], s10 = acc10[v], s11 = acc11[v];
        #pragma unroll
        for (int j = 0; j < 4; ++j) {
          float e0 = As[wr * 32 +  0 + v + 8 * half][k + j];
          float e1 = As[wr * 32 + 16 + v + 8 * half][k + j];
          float f0 = Bs[wc * 32 +  0 + lr][k + j];
          float f1 = Bs[wc * 32 + 16 + lr][k + j];
          s00 += e0 * f0; s01 += e0 * f1; s10 += e1 * f0; s11 += e1 * f1;
        }
        acc00[v] = s00; acc01[v] = s01; acc10[v] = s10; acc11[v] = s11;
      }
#endif
    }
    __syncthreads();
  }

  // ---- epilogue: D layout VGPR v -> M = v + 8*half, N = lr ----
  #pragma unroll
  for (int m = 0; m < 2; ++m) {
    const int rowb = m0 + wr * 32 + m * 16 + half * 8;
    #pragma unroll
    for (int n = 0; n < 2; ++n) {
      const int col = n0 + wc * 32 + n * 16 + lr;
      const v8f* accp = (m == 0) ? (n == 0 ? &acc00 : &acc01)
                                 : (n == 0 ? &acc10 : &acc11);
      #pragma unroll
      for (int v = 0; v < 8; ++v) {
        float x = (*accp)[v];
        if (ADDSRC) x += addsrc[(size_t)(rowb + v) * ldc + col];
        C[(size_t)(rowb + v) * ldc + col] = x;
      }
    }
  }
}

// y[row] = dot(A[row,:], x), one block per row
__global__ __launch_bounds__(256) void ekf_mv_row(
    const float* __restrict__ A, const float* __restrict__ x,
    float* __restrict__ y, int K) {
  __shared__ float red[256];
  const int row = blockIdx.x;
  const float* a = A + (size_t)row * K;
  float sum = 0.f;
  for (int i = threadIdx.x; i < K; i += 256) sum += a[i] * x[i];
  red[threadIdx.x] = sum;
  __syncthreads();
  for (int s = 128; s > 0; s >>= 1) {
    if (threadIdx.x < s) red[threadIdx.x] += red[threadIdx.x + s];
    __syncthreads();
  }
  if (threadIdx.x == 0) y[row] = red[0];
}

// V[m][e] = sum_i B[i][e] * u_m[i]  for m=0..3, single pass over B
__global__ __launch_bounds__(256) void ekf_bt_multi(
    const float* __restrict__ Bmat, int E, int Nn,
    const float* __restrict__ u0, const float* __restrict__ u1,
    const float* __restrict__ u2, const float* __restrict__ u3,
    float* __restrict__ V) {
  const int e = blockIdx.x * 256 + threadIdx.x;
  float v0 = 0.f, v1 = 0.f, v2 = 0.f, v3 = 0.f;
  for (int i = 0; i < Nn; ++i) {
    float b = Bmat[(size_t)i * E + e];
    v0 += b * u0[i]; v1 += b * u1[i]; v2 += b * u2[i]; v3 += b * u3[i];
  }
  V[e] = v0; V[E + e] = v1; V[2 * (size_t)E + e] = v2; V[3 * (size_t)E + e] = v3;
}

// Wc[k][e] = sum_{m=0}^{2-k} a[m+k+1] * V[m][e],  k=0..2
__global__ __launch_bounds__(256) void ekf_wc(
    const float* __restrict__ V, const float* __restrict__ a,
    float* __restrict__ Wc, int E) {
  const int e = blockIdx.x * 256 + threadIdx.x;
  const float a1 = a[1], a2 = a[2], a3 = a[3];
  const float v0 = V[e], v1 = V[E + e], v2 = V[2 * (size_t)E + e];
  Wc[e]                 = a1 * v0 + a2 * v1 + a3 * v2;
  Wc[E + e]             = a2 * v0 + a3 * v1;
  Wc[2 * (size_t)E + e] = a3 * v0;
}

// r[n] = y[n] - (a0*q + a1*u1 + a2*u2 + a3*u3)[n]
__global__ __launch_bounds__(256) void ekf_resid(
    const float* __restrict__ a, const float* __restrict__ q,
    const float* __restrict__ u1, const float* __restrict__ u2,
    const float* __restrict__ u3, const float* __restrict__ y,
    float* __restrict__ r, int Nn) {
  const int n = blockIdx.x * 256 + threadIdx.x;
  if (n < Nn)
    r[n] = y[n] - (a[0] * q[n] + a[1] * u1[n] + a[2] * u2[n] + a[3] * u3[n]);
}

// H[n][e] = B[n][e]*Wc0[e] + P1[n][e]*Wc1[e] + P2[n][e]*Wc2[e]
__global__ __launch_bounds__(256) void ekf_build_h(
    const float* __restrict__ Bmat, const float* __restrict__ P1,
    const float* __restrict__ P2, const float* __restrict__ Wc,
    float* __restrict__ H, int E) {
  const size_t idx = (size_t)blockIdx.x * 256 + threadIdx.x;
  const int e = (int)(idx % (size_t)E);
  H[idx] = Bmat[idx] * Wc[e] + P1[idx] * Wc[E + e] + P2[idx] * Wc[2 * (size_t)E + e];
}

__global__ __launch_bounds__(256) void ekf_transpose(
    const float* __restrict__ S, float* __restrict__ St, int n) {
  const int idx = blockIdx.x * 256 + threadIdx.x;
  const int i = idx / n, j = idx % n;
  St[(size_t)j * n + i] = S[idx];
}

// Single-workgroup Gaussian elimination (no pivoting; S is SPD-like) + back-sub.
// Solves St * z = r for z, St is n x n row-major, n == blockDim.x == 1024.
__global__ __launch_bounds__(1024) void ekf_gauss_solve(
    float* __restrict__ St, const float* __restrict__ r,
    float* __restrict__ z, int n) {
  __shared__ float piv[1024];
  __shared__ float fac[1024];
  __shared__ float rhs[1024];
  const int tid = threadIdx.x;
  rhs[tid] = r[tid];
  __syncthreads();

  for (int k = 0; k < n - 1; ++k) {
    piv[tid] = St[(size_t)k * n + tid];
    __syncthreads();
    const float pinv = 1.0f / piv[k];
    fac[tid] = (tid > k) ? St[(size_t)tid * n + k] * pinv : 0.0f;
    __syncthreads();
    if (tid > k) rhs[tid] -= fac[tid] * rhs[k];
    // column-parallel trailing update (coalesced across tid)
    for (int j = k + 1; j < n; ++j) {
      const float fj = fac[j];
      if (tid > k) St[(size_t)j * n + tid] -= fj * piv[tid];
    }
    __syncthreads();
  }

  // back substitution on upper triangle
  for (int i = n - 1; i >= 0; --i) {
    fac[tid] = (tid > i) ? St[(size_t)i * n + tid] * rhs[tid] : 0.0f;
    __syncthreads();
    for (int s = 512; s > 0; s >>= 1) {
      if (tid < s) fac[tid] += fac[tid + s];
      __syncthreads();
    }
    if (tid == 0) rhs[i] = (rhs[i] - fac[0]) / St[(size_t)i * n + i];
    __syncthreads();
  }
  z[tid] = rhs[tid];
}

// out[e] = max(0, s[e] + sum_n MM[n][e] * z[n])
__global__ __launch_bounds__(256) void ekf_finalize(
    const float* __restrict__ s, const float* __restrict__ MM,
    const float* __restrict__ z, float* __restrict__ out, int Nn, int E) {
  const int e = blockIdx.x * 256 + threadIdx.x;
  float acc = s[e];
  for (int n = 0; n < Nn; ++n) acc += MM[(size_t)n * E + e] * z[n];
  out[e] = fmaxf(acc, 0.0f);
}

extern "C" void kernel_launch(void* const* d_in, const int* in_sizes, int n_in,
                              void* d_out, int out_size, void* d_ws, size_t ws_size,
                              hipStream_t stream) {
  const float* F    = (const float*)d_in[0];  // E x E
  const float* Bmat = (const float*)d_in[1];  // N x E
  const float* C_u  = (const float*)d_in[2];  // E x E
  const float* C_w  = (const float*)d_in[3];  // N x N
  const float* C_x  = (const float*)d_in[4];  // E x E
  const float* s0   = (const float*)d_in[5];  // E
  const float* a    = (const float*)d_in[6];  // 4
  const float* q    = (const float*)d_in[7];  // N
  const float* y    = (const float*)d_in[8];  // N
  float* out = (float*)d_out;

  const int E  = in_sizes[5];   // 4096
  const int Nn = in_sizes[7];   // 1024

  float* ws = (float*)d_ws;
  size_t off = 0;
  float* s  = ws + off; off += (size_t)E;
  float* L  = ws + off; off += (size_t)Nn * Nn;
  float* u1 = ws + off; off += (size_t)Nn;
  float* u2 = ws + off; off += (size_t)Nn;
  float* u3 = ws + off; off += (size_t)Nn;
  float* V  = ws + off; off += 4 * (size_t)E;
  float* Wc = ws + off; off += 3 * (size_t)E;
  float* r  = ws + off; off += (size_t)Nn;
  float* z  = ws + off; off += (size_t)Nn;
  float* P1 = ws + off; off += (size_t)Nn * E;
  float* P2 = ws + off; off += (size_t)Nn * E;
  float* H  = ws + off; off += (size_t)Nn * E;
  float* T1 = ws + off; off += (size_t)Nn * E;   // later reused as MM
  float* T2 = ws + off; off += (size_t)Nn * E;
  float* S  = ws + off; off += (size_t)Nn * Nn;
  float* St = ws + off; off += (size_t)Nn * Nn;

  const dim3 blk(256);
  const dim3 gL(Nn / BN, Nn / BM);   // N-tiles x M-tiles
  const dim3 gP(E / BN, Nn / BM);

  // s = F s0
  ekf_mv_row<<<E, blk, 0, stream>>>(F, s0, s, E);
  // L = (B * diag(s)) B^T
  ekf_gemm_f32_wmma<true, true, false><<<gL, blk, 0, stream>>>(
      Bmat, E, Bmat, E, s, nullptr, L, Nn, Nn, Nn, E);
  // u_m = L^m q
  ekf_mv_row<<<Nn, blk, 0, stream>>>(L, q,  u1, Nn);
  ekf_mv_row<<<Nn, blk, 0, stream>>>(L, u1, u2, Nn);
  ekf_mv_row<<<Nn, blk, 0, stream>>>(L, u2, u3, Nn);
  // V[m] = B^T u_m
  ekf_bt_multi<<<E / 256, blk, 0, stream>>>(Bmat, E, Nn, q, u1, u2, u3, V);
  ekf_wc<<<E / 256, blk, 0, stream>>>(V, a, Wc, E);
  ekf_resid<<<(Nn + 255) / 256, blk, 0, stream>>>(a, q, u1, u2, u3, y, r, Nn);
  // P1 = L B, P2 = L P1
  ekf_gemm_f32_wmma<false, false, false><<<gP, blk, 0, stream>>>(
      L, Nn, Bmat, E, nullptr, nullptr, P1, E, Nn, E, Nn);
  ekf_gemm_f32_wmma<false, false, false><<<gP, blk, 0, stream>>>(
      L, Nn, P1, E, nullptr, nullptr, P2, E, Nn, E, Nn);
  // H = B.*Wc0 + P1.*Wc1 + P2.*Wc2 (broadcast over rows)
  ekf_build_h<<<(int)(((size_t)Nn * E) / 256), blk, 0, stream>>>(Bmat, P1, P2, Wc, H, E);
  // M = H Sigma = ((H F) C_x) F^T + H C_u   (never materialize E x E Sigma)
  ekf_gemm_f32_wmma<false, false, false><<<gP, blk, 0, stream>>>(
      H, E, F, E, nullptr, nullptr, T1, E, Nn, E, E);
  ekf_gemm_f32_wmma<false, false, false><<<gP, blk, 0, stream>>>(
      T1, E, C_x, E, nullptr, nullptr, T2, E, Nn, E, E);
  float* MM = T1;  // T1 dead after previous GEMM; reuse
  ekf_gemm_f32_wmma<true, false, false><<<gP, blk, 0, stream>>>(
      T2, E, F, E, nullptr, nullptr, MM, E, Nn, E, E);
  ekf_gemm_f32_wmma<false, false, true><<<gP, blk, 0, stream>>>(
      H, E, C_u, E, nullptr, MM, MM, E, Nn, E, E);
  // S = M H^T + C_w
  ekf_gemm_f32_wmma<true, false, true><<<gL, blk, 0, stream>>>(
      MM, E, H, E, nullptr, C_w, S, Nn, Nn, Nn, E);
  // solve S^T z = r  (K r = M^T S^{-T} r)
  ekf_transpose<<<(Nn * Nn) / 256, blk, 0, stream>>>(S, St, Nn);
  ekf_gauss_solve<<<1, 1024, 0, stream>>>(St, r, z, Nn);
  // out = relu(s + M^T z)
  ekf_finalize<<<E / 256, blk, 0, stream>>>(s, MM, z, out, Nn, E);
}